// ShufflePatches_45878840656651
// MI455X (gfx1250) — compile-verified
//
#include <hip/hip_runtime.h>

// ShufflePatches: out[b,c, gy*16+r, gx*16+qc] = x[b,c, gyi*16+r, gxi*16+qc]
// where l = perm[b*1024 + gy*32+gx], gyi = l>>5, gxi = l&31.
// Pure data movement: 192 MB total @ 23.3 TB/s => ~8.3 us roofline. Zero FLOPs,
// so WMMA is inapplicable; the CDNA5 feature that applies is the async
// global<->LDS DMA path (ASYNCcnt) + b128 per-lane transfers.

#define AS_GLOBAL __attribute__((address_space(1)))
#define AS_LDS    __attribute__((address_space(3)))

// 16-byte vector pointee type the async builtins expect (per hipcc diagnostic:
// "parameter of type 'int __vector(4) __device__*'").
typedef int v4i __attribute__((vector_size(4 * sizeof(int))));

#if defined(__HIP_DEVICE_COMPILE__) && defined(__gfx1250__) && \
    __has_builtin(__builtin_amdgcn_global_load_async_to_lds_b128) && \
    __has_builtin(__builtin_amdgcn_global_store_async_from_lds_b128)
#define ASYNC_OK 1
#else
#define ASYNC_OK 0
#endif

#if defined(__HIP_DEVICE_COMPILE__) && defined(__gfx1250__) && \
    __has_builtin(__builtin_amdgcn_s_wait_asynccnt)
#define WAIT_ASYNC(n) __builtin_amdgcn_s_wait_asynccnt(n)
#elif defined(__HIP_DEVICE_COMPILE__) && defined(__gfx1250__)
#define WAIT_ASYNC(n) asm volatile("s_wait_asynccnt %0" ::"i"(n) : "memory")
#else
#define WAIT_ASYNC(n)
#endif

// Map a plane-local output quad index (float4 units, 0..65535) to the
// plane-local source quad index via the patch permutation.
__device__ __forceinline__ int src_quad(int pq, int perm_base,
                                        const int* __restrict__ perm) {
    const int q  = pq & 3;          // float4 within patch row (4 per 16-float row)
    const int gx = (pq >> 2) & 31;  // output patch col
    const int h  = pq >> 7;         // output image row (0..511)
    const int r  = h & 15;          // row within patch
    const int gy = h >> 4;          // output patch row
    const int l  = perm[perm_base + (gy << 5) + gx];
    const int gyi = l >> 5;
    const int gxi = l & 31;
    return ((((gyi << 4) + r) << 7) + (gxi << 2) + q);
}

// grid = (32, C=3, B=32), block = 256 (8 waves). Each wave moves 256
// consecutive output float4 (4 KB): fully coalesced b128 stores; 64B-granular
// gathered b128 loads (permutation => every line consumed exactly once; the
// 96MB input fits in the 192MB L2).
extern "C" __global__ __launch_bounds__(256)
void shuffle_patches_kernel(const float4* __restrict__ x,
                            const int* __restrict__ perm,
                            float4* __restrict__ out) {
    const int c    = blockIdx.y;
    const int b    = blockIdx.z;
    const int lane = threadIdx.x & 31;
    const int wave = threadIdx.x >> 5;

    const int  perm_base = b << 10;                    // b * 1024
    const long plane     = (long)(b * 3 + c) << 16;    // * 65536 quads/plane
    const int  chunk     = blockIdx.x * 2048 + wave * 256 + lane;

#if ASYNC_OK
    // Per-wave 4KB staging: memory -> LDS -> memory, no VGPR data registers.
    __shared__ float4 stage[2048];                     // 8 waves * 256 = 32 KB
    const int lds_idx = wave * 256 + lane;

#pragma unroll
    for (int s = 0; s < 8; ++s) {
        const int  pq = chunk + s * 32;
        const long sq = plane + src_quad(pq, perm_base, perm);
        __builtin_amdgcn_global_load_async_to_lds_b128(
            (AS_GLOBAL v4i*)(x + sq),
            (AS_LDS v4i*)(stage + lds_idx + s * 32),
            0, 0);
    }
    WAIT_ASYNC(0);  // all 8 loads landed in LDS (per-wave counter)
#pragma unroll
    for (int s = 0; s < 8; ++s) {
        const int pq = chunk + s * 32;
        __builtin_amdgcn_global_store_async_from_lds_b128(
            (AS_GLOBAL v4i*)(out + plane + pq),
            (AS_LDS v4i*)(stage + lds_idx + s * 32),
            0, 0);
    }
    WAIT_ASYNC(0);  // stores drained before LDS dealloc / wave exit
#else
    // Fallback: direct b128 gather -> coalesced b128 store.
#pragma unroll
    for (int s = 0; s < 8; ++s) {
        const int pq = chunk + s * 32;
        out[plane + pq] = x[plane + src_quad(pq, perm_base, perm)];
    }
#endif
}

extern "C" void kernel_launch(void* const* d_in, const int* in_sizes, int n_in,
                              void* d_out, int out_size, void* d_ws, size_t ws_size,
                              hipStream_t stream) {
    (void)in_sizes; (void)n_in; (void)out_size; (void)d_ws; (void)ws_size;
    const float4* x    = (const float4*)d_in[0];
    const int*    perm = (const int*)d_in[1];
    float4*       out  = (float4*)d_out;

    // 32 blocks per (b,c) plane, planes on (y,z). 256 threads = 8 wave32.
    dim3 grid(32, 3, 32);
    dim3 block(256, 1, 1);
    shuffle_patches_kernel<<<grid, block, 0, stream>>>(x, perm, out);
}